// HierarchicalAttention_62380105007262
// MI455X (gfx1250) — compile-verified
//
#include <hip/hip_runtime.h>

typedef __attribute__((ext_vector_type(2))) float v2f;
typedef __attribute__((ext_vector_type(8))) float v8f;

#define MEM_N   512
#define BATCH   64
#define QLEN    2048
#define NBITS   9

// ---------------------------------------------------------------------------
// Kernel A: table[b][a] = invZ * sum_n memory[b][n] * exp(-2*popcount(n^a))
// Computed as a 64x512 @ 512x512 GEMM with V_WMMA_F32_16X16X4_F32 (exact f32).
// One wave32 per 16x16 output tile; 128 k-steps of K=4.
// A-operand layout (ISA 7.12.2, 32-bit A 16x4): lane = M%16 within each half;
//   lanes 0-15 hold K=k0+0 (x) / k0+1 (y), lanes 16-31 hold K=k0+2 / k0+3.
// B-operand (4x16, K x N) mirrored: lane%16 = N column, halves split K 0,1/2,3.
// C/D layout: VGPR r holds M = r + 8*(lane>=16), N = lane%16.
// ---------------------------------------------------------------------------
__global__ __launch_bounds__(256) void build_table_wmma(
    const float* __restrict__ memory, float* __restrict__ table) {
  const int lane = threadIdx.x & 31;
  const int wave = blockIdx.x * (blockDim.x >> 5) + (threadIdx.x >> 5);

  // 4 row tiles (batch) x 32 col tiles (address) = 128 tiles total
  const int tb = wave >> 5;   // 0..3
  const int ta = wave & 31;   // 0..31
  const int b0 = tb * 16;
  const int a0 = ta * 16;

  const int half = lane >> 4;   // 0 or 1
  const int l15  = lane & 15;

  const float* arow = memory + (size_t)(b0 + l15) * MEM_N;  // A row: M = lane%16
  const int col = a0 + l15;                                  // B col: N = lane%16

  v8f acc = {};
#pragma unroll 4
  for (int k0 = 0; k0 < MEM_N; k0 += 4) {
    const int k = k0 + half * 2;

    v2f a;
    a.x = arow[k];
    a.y = arow[k + 1];

    v2f b;
    b.x = __expf(-2.0f * (float)__popc(k ^ col));
    b.y = __expf(-2.0f * (float)__popc((k + 1) ^ col));

    // (neg_a, A, neg_b, B, c_mod, C, reuse_a, reuse_b)
    acc = __builtin_amdgcn_wmma_f32_16x16x4_f32(false, a, false, b,
                                                (short)0, acc, false, false);
  }

  // softmax denominator is address-independent: Z = (1+e^-2)^9
  const float invZ = 0.3190667463f;

#pragma unroll
  for (int r = 0; r < 8; ++r) {
    const int m = b0 + r + half * 8;
    table[(size_t)m * MEM_N + col] = acc[r] * invZ;
  }
}

// ---------------------------------------------------------------------------
// Kernel B: out[b][q] = table[b][query_addr[b][q]]
// Pure gather; table (128KB) is L2-resident after kernel A.
// ---------------------------------------------------------------------------
__global__ __launch_bounds__(256) void gather_out(
    const int* __restrict__ qaddr, const float* __restrict__ table,
    float* __restrict__ out, int total) {
  int i = blockIdx.x * blockDim.x + threadIdx.x;
  if (i < total) {
    const int b = i >> 11;                 // QLEN = 2048
    const int a = qaddr[i] & (MEM_N - 1);  // guard (addresses are 0..511)
    out[i] = table[(size_t)b * MEM_N + a];
  }
}

extern "C" void kernel_launch(void* const* d_in, const int* in_sizes, int n_in,
                              void* d_out, int out_size, void* d_ws, size_t ws_size,
                              hipStream_t stream) {
  const int*   qaddr  = (const int*)d_in[0];    // query_addr [64, 2048] int32
  const float* memory = (const float*)d_in[1];  // memory     [64, 512]  f32
  float*       out    = (float*)d_out;          // [64, 2048] f32
  float*       table  = (float*)d_ws;           // [64, 512]  f32 scratch (128 KB)

  // 128 tiles, 8 waves per 256-thread block -> 16 blocks
  build_table_wmma<<<16, 256, 0, stream>>>(memory, table);

  const int total = BATCH * QLEN;  // 131072
  gather_out<<<(total + 255) / 256, 256, 0, stream>>>(qaddr, table, out, total);
}